// SimpleLatentProto_54460185313529
// MI455X (gfx1250) — compile-verified
//
#include <hip/hip_runtime.h>
#include <cmath>

typedef __attribute__((ext_vector_type(2))) float v2f;
typedef __attribute__((ext_vector_type(8))) float v8f;

#define IN_DIM 512
#define EPS_N  1e-12f
#define NT     4          // 16-col WMMA tiles per wave per pass
#define DPF    4          // B prefetch pipeline depth (k-steps)

// ---- normalize weight rows -> paired-K layout wnP[(k/2)*OUT + n][2], w2[OUT]
__global__ __launch_bounds__(256) void norm_rows_w(
    const float* __restrict__ w, float* __restrict__ wnP,
    float* __restrict__ w2, int outDim) {
  const int o = blockIdx.x;
  const int t = threadIdx.x;            // 256 threads, k = 2t, 2t+1
  const float a0 = w[(size_t)o * IN_DIM + 2 * t];
  const float a1 = w[(size_t)o * IN_DIM + 2 * t + 1];
  float s = a0 * a0 + a1 * a1;
#pragma unroll
  for (int off = 16; off; off >>= 1) s += __shfl_xor(s, off, 32);
  __shared__ float red[8];
  __shared__ float tot;
  if ((t & 31) == 0) red[t >> 5] = s;
  __syncthreads();
  if (t == 0) {
    float v = 0.0f;
#pragma unroll
    for (int i = 0; i < 8; ++i) v += red[i];
    tot = v;
  }
  __syncthreads();
  const float ss  = tot;
  const float inv = 1.0f / fmaxf(sqrtf(ss), EPS_N);
  if (t == 0) w2[o] = ss * inv * inv;
  float2 pr;
  pr.x = a0 * inv;                      // K = 2t
  pr.y = a1 * inv;                      // K = 2t+1
  ((float2*)wnP)[(size_t)t * outDim + o] = pr;   // one contiguous 8B store
}

// ---------------- normalize x rows -> xn[B][IN] row-major, x2[B] -------------
__global__ __launch_bounds__(256) void norm_rows_x(
    const float* __restrict__ x, float* __restrict__ xn, float* __restrict__ x2) {
  const int b = blockIdx.x;
  const int t = threadIdx.x;
  const float a0 = x[(size_t)b * IN_DIM + 2 * t];
  const float a1 = x[(size_t)b * IN_DIM + 2 * t + 1];
  float s = a0 * a0 + a1 * a1;
#pragma unroll
  for (int off = 16; off; off >>= 1) s += __shfl_xor(s, off, 32);
  __shared__ float red[8];
  __shared__ float tot;
  if ((t & 31) == 0) red[t >> 5] = s;
  __syncthreads();
  if (t == 0) {
    float v = 0.0f;
#pragma unroll
    for (int i = 0; i < 8; ++i) v += red[i];
    tot = v;
  }
  __syncthreads();
  const float ss  = tot;
  const float inv = 1.0f / fmaxf(sqrtf(ss), EPS_N);
  if (t == 0) x2[b] = ss * inv * inv;
  xn[(size_t)b * IN_DIM + 2 * t]     = a0 * inv;
  xn[(size_t)b * IN_DIM + 2 * t + 1] = a1 * inv;
}

// ---- fused: 16-row logits strip via WMMA f32, sparsemax in 320KB LDS -------
__global__ __launch_bounds__(256) void fused_gemm_sparsemax(
    const float* __restrict__ xn, const float* __restrict__ x2,
    const float* __restrict__ wnP, const float* __restrict__ w2,
    const float* __restrict__ lambd, float* __restrict__ out, int outDim) {
  extern __shared__ float lds[];
  const int zstride = outDim + 4;        // bank-conflict padding (+4 floats per row)
  float* Alds = lds;                     // [16][IN_DIM]   = 32 KB
  float* Z    = lds + 16 * IN_DIM;       // [16][zstride]  ~ 256 KB
  __shared__ float X2s[16];

  const int t  = threadIdx.x;
  const int m0 = blockIdx.x * 16;

  if (t < 16) X2s[t] = x2[m0 + t];

  {  // cooperative A-tile load (16 x 512 fp32) into LDS
    const float4* s4 = (const float4*)(xn + (size_t)m0 * IN_DIM);
    float4* d4 = (float4*)Alds;
    for (int i = t; i < (16 * IN_DIM) / 4; i += 256) d4[i] = s4[i];
  }
  __syncthreads();

  const float lam  = lambd[0];
  const float cpos = 1.0f + 2.0f * lam;

  const int wave = t >> 5;
  const int lane = t & 31;
  const int half = lane >> 4;            // K-pair select / M-half (ISA layout)
  const int ln   = lane & 15;

  const int colsPerWave = outDim >> 3;            // 512
  const int passes      = colsPerWave / (NT * 16);// 8
  const int nwbase      = wave * colsPerWave;

  const float* Arow = Alds + ln * IN_DIM;         // A lane row m = ln
  const uint32_t kstep = 4u * (uint32_t)outDim;   // float offset per k-step

  for (int pass = 0; pass < passes; ++pass) {
    const int n0 = nwbase + pass * (NT * 16);
    v8f acc[NT];
#pragma unroll
    for (int tI = 0; tI < NT; ++tI) acc[tI] = (v8f){};

    // base float offset of this lane's B pair: ((k>>1)+half)*2*outDim + 2n
    const uint32_t off = (uint32_t)half * 2u * (uint32_t)outDim +
                         2u * (uint32_t)(n0 + ln);

    v2f bbuf[DPF][NT];                            // register prefetch pipeline
#pragma unroll
    for (int p = 0; p < DPF; ++p)
#pragma unroll
      for (int tI = 0; tI < NT; ++tI)
        bbuf[p][tI] = *(const v2f*)(wnP + off + (uint32_t)p * kstep + 32u * tI);

    for (int kb = 0; kb < (IN_DIM / 4) / DPF; ++kb) {  // 32 outer iterations
#pragma unroll
      for (int u = 0; u < DPF; ++u) {
        const int ks = kb * DPF + u;
        v2f a;
        a.x = Arow[4 * ks + 2 * half];
        a.y = Arow[4 * ks + 2 * half + 1];
        v2f bcur[NT];
#pragma unroll
        for (int tI = 0; tI < NT; ++tI) bcur[tI] = bbuf[u][tI];
        // prefetch k-step ks+DPF (tail overreads into adjacent workspace; unused)
        const uint32_t poff = off + (uint32_t)(ks + DPF) * kstep;
#pragma unroll
        for (int tI = 0; tI < NT; ++tI)
          bbuf[u][tI] = *(const v2f*)(wnP + poff + 32u * tI);
#pragma unroll
        for (int tI = 0; tI < NT; ++tI)
          acc[tI] = __builtin_amdgcn_wmma_f32_16x16x4_f32(
              false, a, false, bcur[tI], (short)0, acc[tI], false, false);
      }
    }

    // logits = (1+2λ)·xa − λ·(x2[m] + w2[n]); store strip into LDS
#pragma unroll
    for (int tI = 0; tI < NT; ++tI) {
      const int nc = n0 + 16 * tI + ln;
      const float w2n = w2[nc];
#pragma unroll
      for (int v = 0; v < 8; ++v) {
        const int m = v + (half << 3);            // C/D layout: row = v + 8*half
        Z[m * zstride + nc] = cpos * acc[tI][v] - lam * (X2s[m] + w2n);
      }
    }
  }
  __syncthreads();

  // ---- sparsemax: 16 lanes own one row; bisection on tau, float4 LDS reads --
  const int r = t >> 4;
  const int j = t & 15;
  const float4* zr4 = (const float4*)(Z + r * zstride);
  const int nq = outDim >> 2;                     // 1024 float4 per row
  float mx = -3.4e38f;
  for (int i = j; i < nq; i += 16) {
    const float4 q = zr4[i];
    mx = fmaxf(mx, fmaxf(fmaxf(q.x, q.y), fmaxf(q.z, q.w)));
  }
#pragma unroll
  for (int off = 8; off; off >>= 1) mx = fmaxf(mx, __shfl_xor(mx, off, 16));
  float lo  = mx - 1.0f;                          // tau* in [max-1, max]
  float hiB = mx;
  for (int iter = 0; iter < 30; ++iter) {
    const float tau = 0.5f * (lo + hiB);
    float s = 0.0f;
    for (int i = j; i < nq; i += 16) {
      const float4 q = zr4[i];
      s += fmaxf(q.x - tau, 0.0f) + fmaxf(q.y - tau, 0.0f) +
           fmaxf(q.z - tau, 0.0f) + fmaxf(q.w - tau, 0.0f);
    }
#pragma unroll
    for (int off = 8; off; off >>= 1) s += __shfl_xor(s, off, 16);
    if (s >= 1.0f) lo = tau; else hiB = tau;
  }
  const float tau = 0.5f * (lo + hiB);
  float4* orow = (float4*)(out + (size_t)(m0 + r) * outDim);
  for (int i = j; i < nq; i += 16) {
    const float4 q = zr4[i];
    float4 o;
    o.x = fmaxf(q.x - tau, 0.0f);
    o.y = fmaxf(q.y - tau, 0.0f);
    o.z = fmaxf(q.z - tau, 0.0f);
    o.w = fmaxf(q.w - tau, 0.0f);
    orow[i] = o;
  }
}

extern "C" void kernel_launch(void* const* d_in, const int* in_sizes, int n_in,
                              void* d_out, int out_size, void* d_ws, size_t ws_size,
                              hipStream_t stream) {
  const float* x     = (const float*)d_in[0];
  const float* w     = (const float*)d_in[1];
  const float* lambd = (const float*)d_in[2];
  float* out = (float*)d_out;

  const int B   = in_sizes[0] / IN_DIM;   // 8192
  const int OUT = in_sizes[1] / IN_DIM;   // 4096

  float* ws  = (float*)d_ws;
  float* wnP = ws;                               // IN*OUT  (8 MB, paired-K)
  float* w2  = wnP + (size_t)IN_DIM * OUT;       // OUT
  float* xn  = w2 + OUT;                         // B*IN    (16 MB)
  float* x2  = xn + (size_t)B * IN_DIM;          // B

  norm_rows_w<<<OUT, 256, 0, stream>>>(w, wnP, w2, OUT);
  norm_rows_x<<<B, 256, 0, stream>>>(x, xn, x2);

  const size_t ldsBytes =
      (size_t)(16 * IN_DIM + 16 * (OUT + 4)) * sizeof(float);  // ~288.25 KB
  hipFuncSetAttribute((const void*)fused_gemm_sparsemax,
                      hipFuncAttributeMaxDynamicSharedMemorySize, (int)ldsBytes);
  fused_gemm_sparsemax<<<B / 16, 256, ldsBytes, stream>>>(xn, x2, wnP, w2,
                                                          lambd, out, OUT);
}